// JPEGCompression_4655744549002
// MI455X (gfx1250) — compile-verified
//
#include <hip/hip_runtime.h>

// ---------------------------------------------------------------------------
// JPEG round-trip (quality 95, 4:2:0) fused kernel for gfx1250.
// 8x8 DCT/IDCT done as 16x16 block-diagonal matmuls on V_WMMA_F32_16X16X4_F32.
// One wave32 per 32x32 pixel region; 6 tiles/region * 16 WMMA/tile.
// This rev: fully constant-folded quant tables + reciprocals (no runtime
// divides/floors in table setup).
// ---------------------------------------------------------------------------

typedef float v2f __attribute__((ext_vector_type(2)));
typedef float v8f __attribute__((ext_vector_type(8)));

#define LSTRIDE 33   // padded 32x32 luma plane stride
#define TSTRIDE 17   // padded 16x16 tile stride

// DCT-II basis D[k][i] = sqrt(2/8)*cos((2i+1)k*pi/16), row 0 scaled 1/sqrt(2)
__device__ __constant__ float kD[64] = {
  0.35355339059327373f,  0.35355339059327373f,  0.35355339059327373f,  0.35355339059327373f,
  0.35355339059327373f,  0.35355339059327373f,  0.35355339059327373f,  0.35355339059327373f,
  0.49039264020161522f,  0.41573480615127262f,  0.27778511650980114f,  0.09754516100806417f,
 -0.09754516100806417f, -0.27778511650980114f, -0.41573480615127262f, -0.49039264020161522f,
  0.46193976625564337f,  0.19134171618254492f, -0.19134171618254492f, -0.46193976625564337f,
 -0.46193976625564337f, -0.19134171618254492f,  0.19134171618254492f,  0.46193976625564337f,
  0.41573480615127262f, -0.09754516100806417f, -0.49039264020161522f, -0.27778511650980114f,
  0.27778511650980114f,  0.49039264020161522f,  0.09754516100806417f, -0.41573480615127262f,
  0.35355339059327373f, -0.35355339059327373f, -0.35355339059327373f,  0.35355339059327373f,
  0.35355339059327373f, -0.35355339059327373f, -0.35355339059327373f,  0.35355339059327373f,
  0.27778511650980114f, -0.49039264020161522f,  0.09754516100806417f,  0.41573480615127262f,
 -0.41573480615127262f, -0.09754516100806417f,  0.49039264020161522f, -0.27778511650980114f,
  0.19134171618254492f, -0.46193976625564337f,  0.46193976625564337f, -0.19134171618254492f,
 -0.19134171618254492f,  0.46193976625564337f, -0.46193976625564337f,  0.19134171618254492f,
  0.09754516100806417f, -0.27778511650980114f,  0.41573480615127262f, -0.49039264020161522f,
  0.49039264020161522f, -0.41573480615127262f,  0.27778511650980114f, -0.09754516100806417f };

// Quality-95 quant tables: q = clip(floor((base*10+50)/100), 1, 255),
// precomputed, with reciprocals as compile-time-folded literals.
__device__ __constant__ float kQL[64] = {
  2,1,1,2,2,4,5,6,
  1,1,1,2,3,6,6,6,
  1,1,2,2,4,6,7,6,
  1,2,2,3,5,9,8,6,
  2,2,4,6,7,11,10,8,
  2,4,6,6,8,10,11,9,
  5,6,8,9,10,12,12,10,
  7,9,10,10,11,10,10,10 };

__device__ __constant__ float kQLr[64] = {
  1.f/2,1.f/1,1.f/1,1.f/2,1.f/2,1.f/4,1.f/5,1.f/6,
  1.f/1,1.f/1,1.f/1,1.f/2,1.f/3,1.f/6,1.f/6,1.f/6,
  1.f/1,1.f/1,1.f/2,1.f/2,1.f/4,1.f/6,1.f/7,1.f/6,
  1.f/1,1.f/2,1.f/2,1.f/3,1.f/5,1.f/9,1.f/8,1.f/6,
  1.f/2,1.f/2,1.f/4,1.f/6,1.f/7,1.f/11,1.f/10,1.f/8,
  1.f/2,1.f/4,1.f/6,1.f/6,1.f/8,1.f/10,1.f/11,1.f/9,
  1.f/5,1.f/6,1.f/8,1.f/9,1.f/10,1.f/12,1.f/12,1.f/10,
  1.f/7,1.f/9,1.f/10,1.f/10,1.f/11,1.f/10,1.f/10,1.f/10 };

__device__ __constant__ float kQC[64] = {
  2,2,2,5,10,10,10,10,
  2,2,3,7,10,10,10,10,
  2,3,6,10,10,10,10,10,
  5,7,10,10,10,10,10,10,
  10,10,10,10,10,10,10,10,
  10,10,10,10,10,10,10,10,
  10,10,10,10,10,10,10,10,
  10,10,10,10,10,10,10,10 };

__device__ __constant__ float kQCr[64] = {
  1.f/2,1.f/2,1.f/2,1.f/5,1.f/10,1.f/10,1.f/10,1.f/10,
  1.f/2,1.f/2,1.f/3,1.f/7,1.f/10,1.f/10,1.f/10,1.f/10,
  1.f/2,1.f/3,1.f/6,1.f/10,1.f/10,1.f/10,1.f/10,1.f/10,
  1.f/5,1.f/7,1.f/10,1.f/10,1.f/10,1.f/10,1.f/10,1.f/10,
  1.f/10,1.f/10,1.f/10,1.f/10,1.f/10,1.f/10,1.f/10,1.f/10,
  1.f/10,1.f/10,1.f/10,1.f/10,1.f/10,1.f/10,1.f/10,1.f/10,
  1.f/10,1.f/10,1.f/10,1.f/10,1.f/10,1.f/10,1.f/10,1.f/10,
  1.f/10,1.f/10,1.f/10,1.f/10,1.f/10,1.f/10,1.f/10,1.f/10 };

// Block-diagonal 16x16 basis: diag(D, D)
__device__ __forceinline__ float ddiag(int m, int k) {
  const float v = kD[(m & 7) * 8 + (k & 7)];
  return ((m >> 3) == (k >> 3)) ? v : 0.0f;
}

// acc = A * src^T, A given as per-lane 16x4 fragments (4 K-chunks),
// src is a 16x16 f32 matrix in LDS with row stride sstride.
// B-fragment of src^T: B[k][n] = src[n][k].
__device__ __forceinline__ v8f wmma_AxBT(const v2f a[4], const float* src,
                                         int sstride) {
  const int lane = threadIdx.x & 31;
  const int n    = lane & 15;
  const int kb   = (lane >> 4) << 1;   // 0 or 2
  v8f acc = {0.f, 0.f, 0.f, 0.f, 0.f, 0.f, 0.f, 0.f};
#pragma unroll
  for (int kc = 0; kc < 4; ++kc) {
    const int k0 = 4 * kc + kb;
    v2f b;
    b.x = src[n * sstride + k0];
    b.y = src[n * sstride + k0 + 1];
    acc = __builtin_amdgcn_wmma_f32_16x16x4_f32(
        /*neg_a=*/false, a[kc], /*neg_b=*/false, b,
        /*c_mod=*/(short)0, acc, /*reuse_a=*/false, /*reuse_b=*/false);
  }
  return acc;
}

// Scatter a 16x16 f32 C/D fragment (8 VGPRs) to LDS.
// C layout: VGPR r holds row m = r + (lane>>4)*8, col n = lane&15.
__device__ __forceinline__ void storeC(float* dst, int dstride, v8f c) {
  const int lane = threadIdx.x & 31;
  const int n    = lane & 15;
  const int m0   = (lane >> 4) << 3;
#pragma unroll
  for (int r = 0; r < 8; ++r) dst[(m0 + r) * dstride + n] = c[r];
}

// Full DCT -> quant -> dequant -> IDCT on one 16x16 tile (4 DCT blocks).
// tile: level-shifted samples, in/out, in LDS. scr: 16x16 LDS scratch.
// qv/rqv: this lane's quant column (q and 1/q for rows 0..7 at col lane&7).
// Chain (A = Ddiag, At = Ddiag^T):
//   T1 = A*X^T;  T2 = A*T1^T = coef;  quantize;
//   T3 = At*T2^T; T4 = At*T3^T = rec.
__device__ __forceinline__ void jpeg_tile(float* tile, int tstride, float* scr,
                                          const v2f aF[4], const v2f aI[4],
                                          const float qv[8], const float rqv[8]) {
  v8f t;
  t = wmma_AxBT(aF, tile, tstride);
  storeC(scr, TSTRIDE, t);
  t = wmma_AxBT(aF, scr, TSTRIDE);          // t = coef (C layout)
#pragma unroll
  for (int r = 0; r < 8; ++r)               // q row index = m&7 = r
    t[r] = rintf(t[r] * rqv[r]) * qv[r];    // round half-even like jnp.round
  storeC(scr, TSTRIDE, t);
  t = wmma_AxBT(aI, scr, TSTRIDE);
  storeC(scr, TSTRIDE, t);
  t = wmma_AxBT(aI, scr, TSTRIDE);          // t = rec
  storeC(tile, tstride, t);
}

extern "C" __global__ __launch_bounds__(32)
void jpeg_roundtrip_kernel(const float* __restrict__ x, float* __restrict__ out) {
  __shared__ float Yp [32 * LSTRIDE];       // level-shifted luma, 32x32
  __shared__ float Cbd[16 * TSTRIDE];       // downsampled level-shifted Cb
  __shared__ float Crd[16 * TSTRIDE];
  __shared__ float Scr[16 * TSTRIDE];

  const int lane = threadIdx.x;
  const int b    = blockIdx.z;
  const int row0 = blockIdx.y * 32;
  const int col0 = blockIdx.x * 32;

  // ---- load 32x32 RGB, color convert (level-shifted), downsample chroma ----
  const float* px = x + ((size_t)b * (512 * 512) + (size_t)row0 * 512 + col0) * 3;
  float cbPrev = 0.0f, crPrev = 0.0f;
  for (int r = 0; r < 32; ++r) {
    const float* p = px + ((size_t)r * 512 + lane) * 3;
    const float rr = fminf(fmaxf(floorf(p[0] * 255.0f), 0.0f), 255.0f);
    const float gg = fminf(fmaxf(floorf(p[1] * 255.0f), 0.0f), 255.0f);
    const float bb = fminf(fmaxf(floorf(p[2] * 255.0f), 0.0f), 255.0f);
    const float y  =  0.299f    * rr + 0.587f    * gg + 0.114f * bb - 128.0f;
    const float cb = -0.168736f * rr - 0.331264f * gg + 0.5f   * bb;    // cb-128
    const float cr =  0.5f      * rr - 0.418688f * gg - 0.081312f * bb; // cr-128
    Yp[r * LSTRIDE + lane] = y;
    const float cbh = cb + __shfl_xor(cb, 1, 32);   // horizontal 2:1 sum
    const float crh = cr + __shfl_xor(cr, 1, 32);
    if (r & 1) {
      if ((lane & 1) == 0) {
        Cbd[(r >> 1) * TSTRIDE + (lane >> 1)] = (cbPrev + cbh) * 0.25f;
        Crd[(r >> 1) * TSTRIDE + (lane >> 1)] = (crPrev + crh) * 0.25f;
      }
    } else {
      cbPrev = cbh;
      crPrev = crh;
    }
  }

  // ---- per-lane A fragments of Ddiag (aF) and Ddiag^T (aI) ----
  // A 16x4 f32 layout: lane m = lane&15; VGPR v holds K = (lane>>4)*2 + v.
  v2f aF[4], aI[4];
  {
    const int mA = lane & 15;
    const int kb = (lane >> 4) << 1;
#pragma unroll
    for (int kc = 0; kc < 4; ++kc) {
      const int k0 = 4 * kc + kb;
      aF[kc].x = ddiag(mA, k0);
      aF[kc].y = ddiag(mA, k0 + 1);
      aI[kc].x = ddiag(k0, mA);        // Ddiag^T[m][k] = Ddiag[k][m]
      aI[kc].y = ddiag(k0 + 1, mA);
    }
  }

  // ---- per-lane quant columns: straight gathers from constant tables ----
  float qlv[8], qlr[8], qcv[8], qcr[8];
  {
    const int qc = lane & 7;
#pragma unroll
    for (int r = 0; r < 8; ++r) {
      qlv[r] = kQL [r * 8 + qc];
      qlr[r] = kQLr[r * 8 + qc];
      qcv[r] = kQC [r * 8 + qc];
      qcr[r] = kQCr[r * 8 + qc];
    }
  }

  // ---- luma: four 16x16 tiles (each = 2x2 DCT blocks) ----
  for (int ty = 0; ty < 2; ++ty)
    for (int tx = 0; tx < 2; ++tx)
      jpeg_tile(&Yp[(ty * 16) * LSTRIDE + tx * 16], LSTRIDE, Scr, aF, aI, qlv, qlr);

  // ---- chroma: one 16x16 tile each ----
  jpeg_tile(Cbd, TSTRIDE, Scr, aF, aI, qcv, qcr);
  jpeg_tile(Crd, TSTRIDE, Scr, aF, aI, qcv, qcr);

  // ---- recombine, clamp, round, scale back to [0,1] ----
  float* po = out + ((size_t)b * (512 * 512) + (size_t)row0 * 512 + col0) * 3;
  for (int r = 0; r < 32; ++r) {
    const float y2  = Yp[r * LSTRIDE + lane] + 128.0f;
    const float cb2 = Cbd[(r >> 1) * TSTRIDE + (lane >> 1)];  // = cb2-128 upsampled
    const float cr2 = Crd[(r >> 1) * TSTRIDE + (lane >> 1)];
    const float r2 = y2 + 1.402f    * cr2;
    const float g2 = y2 - 0.344136f * cb2 - 0.714136f * cr2;
    const float b2 = y2 + 1.772f    * cb2;
    float* q = po + ((size_t)r * 512 + lane) * 3;
    q[0] = rintf(fminf(fmaxf(r2, 0.0f), 255.0f)) * (1.0f / 255.0f);
    q[1] = rintf(fminf(fmaxf(g2, 0.0f), 255.0f)) * (1.0f / 255.0f);
    q[2] = rintf(fminf(fmaxf(b2, 0.0f), 255.0f)) * (1.0f / 255.0f);
  }
}

extern "C" void kernel_launch(void* const* d_in, const int* in_sizes, int n_in,
                              void* d_out, int out_size, void* d_ws, size_t ws_size,
                              hipStream_t stream) {
  (void)in_sizes; (void)n_in; (void)d_ws; (void)ws_size; (void)out_size;
  const float* x = (const float*)d_in[0];
  float* out = (float*)d_out;
  dim3 grid(512 / 32, 512 / 32, 32);   // 16 x 16 x 32 = 8192 single-wave blocks
  jpeg_roundtrip_kernel<<<grid, dim3(32, 1, 1), 0, stream>>>(x, out);
}